// MoD_17703855194814
// MI455X (gfx1250) — compile-verified
//
#include <hip/hip_runtime.h>

typedef __attribute__((ext_vector_type(2))) float v2f;
typedef __attribute__((ext_vector_type(8))) float v8f;

#define B_  4
#define S_  4096
#define D_  1024
#define K_  2048   // top_k

// 32-bit LDS byte offset of a __shared__ object (real addrspacecast -> as3)
#define LDS_OFF(p) ((unsigned)(unsigned long long)(__attribute__((address_space(3))) void*)(p))

// ---------------------------------------------------------------------------
// Kernel 1: router logits. One wave per token row; float4 loads, shfl reduce.
// ---------------------------------------------------------------------------
__global__ void __launch_bounds__(256) mod_router(const float* __restrict__ x,
                                                  const float* __restrict__ Wr,
                                                  float* __restrict__ logits) {
    int row  = (blockIdx.x * blockDim.x + threadIdx.x) >> 5;   // 0 .. B*S-1
    int lane = threadIdx.x & 31;
    const float4* xr = (const float4*)(x + (size_t)row * D_);
    const float4* wr = (const float4*)Wr;
    float acc = 0.f;
#pragma unroll
    for (int i = 0; i < D_ / 4 / 32; ++i) {
        float4 a = xr[i * 32 + lane];
        float4 w = wr[i * 32 + lane];
        acc += a.x * w.x + a.y * w.y + a.z * w.z + a.w * w.w;
    }
#pragma unroll
    for (int off = 16; off; off >>= 1) acc += __shfl_xor(acc, off, 32);
    if (lane == 0) logits[row] = acc;
}

// ---------------------------------------------------------------------------
// Kernel 2: exact top-K via total-order rank (no atomics, deterministic).
// ---------------------------------------------------------------------------
__global__ void __launch_bounds__(256) mod_select(const float* __restrict__ logits,
                                                  int* __restrict__ idx,
                                                  unsigned char* __restrict__ flags) {
    __shared__ float L[S_];
    int b = blockIdx.y;
    const float* lb = logits + b * S_;
    for (int i = threadIdx.x; i < S_; i += 256) L[i] = lb[i];
    __syncthreads();

    int s = blockIdx.x * 256 + threadIdx.x;
    float my = L[s];
    int cnt = 0;
#pragma unroll 8
    for (int t = 0; t < S_; ++t) {
        float v = L[t];
        cnt += (v > my) || (v == my && t < s);
    }
    if (cnt < K_) {
        idx[b * K_ + cnt] = s;
        flags[b * S_ + s] = 1;
    } else {
        flags[b * S_ + s] = 0;
    }
}

// ---------------------------------------------------------------------------
// Kernel 3: copy un-selected rows x -> out (float4).
// ---------------------------------------------------------------------------
__global__ void __launch_bounds__(64) mod_copy(const float* __restrict__ x,
                                               const unsigned char* __restrict__ flags,
                                               float* __restrict__ out) {
    int row = blockIdx.x;
    if (flags[row]) return;
    const float4* src = (const float4*)(x + (size_t)row * D_);
    float4*       dst = (float4*)(out + (size_t)row * D_);
#pragma unroll
    for (int i = 0; i < 4; ++i) dst[threadIdx.x + 64 * i] = src[threadIdx.x + 64 * i];
}

// ---------------------------------------------------------------------------
// Kernel 4: WMMA fp32 GEMM, double-buffered async-to-LDS pipeline.
// Block tile 128(M) x 256(N); K in steps of 32; 8 waves, 64x64 C per wave.
// Per tile each wave issues 12 async B128s; next tile is prefetched while the
// current one is consumed (s_wait_asynccnt 12 = "previous tile resident").
// ---------------------------------------------------------------------------
#define LDA 36
#define LDB 36

__global__ void __launch_bounds__(256) mod_gemm(const float* __restrict__ x,
                                                const float* __restrict__ Wb,
                                                const int* __restrict__ idx,
                                                float* __restrict__ out) {
    __shared__ float As[2][128][LDA];   // As[buf][m][k]
    __shared__ float Bs[2][256][LDB];   // Bs[buf][n][k] = Wb[n0+n][k0+k]
    __shared__ int   rowTok[128];

    int tid = threadIdx.x;
    int m0  = blockIdx.x * 128;         // over B*K = 8192 gathered rows
    int n0  = blockIdx.y * 256;         // over D = 1024 output features
    int b   = m0 >> 11;                 // 2048 rows/batch; 2048 % 128 == 0

    if (tid < 128) rowTok[tid] = idx[b * K_ + (m0 & (K_ - 1)) + tid];
    __syncthreads();

    int wave = tid >> 5, lane = tid & 31;
    int wm = wave & 1, wn = wave >> 1;              // 2 x 4 wave grid
    int half = lane >> 4, l16 = lane & 15;

    // staging geometry: 8 threads per row, 16B chunks
    const int tr = tid >> 3;                        // 0..31
    const int cx = (tid & 7) * 4;                   // 0..28

    unsigned abase[4], bbase[8];
#pragma unroll
    for (int st = 0; st < 4; ++st)
        abase[st] = (((unsigned)b * S_ + (unsigned)rowTok[st * 32 + tr]) * D_ + cx) * 4u;
#pragma unroll
    for (int st = 0; st < 8; ++st)
        bbase[st] = (((unsigned)(n0 + st * 32 + tr)) * D_ + cx) * 4u;

    auto issue_tile = [&](int k0, int buf) {
#pragma unroll
        for (int st = 0; st < 4; ++st) {
            unsigned goff = abase[st] + (unsigned)k0 * 4u;
            unsigned loff = LDS_OFF(&As[buf][st * 32 + tr][cx]);
            asm volatile("global_load_async_to_lds_b128 %0, %1, %2"
                         :: "v"(loff), "v"(goff), "s"(x) : "memory");
        }
#pragma unroll
        for (int st = 0; st < 8; ++st) {
            unsigned goff = bbase[st] + (unsigned)k0 * 4u;
            unsigned loff = LDS_OFF(&Bs[buf][st * 32 + tr][cx]);
            asm volatile("global_load_async_to_lds_b128 %0, %1, %2"
                         :: "v"(loff), "v"(goff), "s"(Wb) : "memory");
        }
    };

    v8f c[4][4] = {};                   // 64x64 accumulator per wave

    issue_tile(0, 0);                   // prologue: tile 0 in flight

    for (int k0 = 0; k0 < D_; k0 += 32) {
        int  buf  = (k0 >> 5) & 1;
        bool more = (k0 + 32) < D_;

        if (more) {
            issue_tile(k0 + 32, buf ^ 1);            // prefetch next tile
            asm volatile("s_wait_asynccnt 12" ::: "memory");   // tile k0 resident
        } else {
            asm volatile("s_wait_asynccnt 0" ::: "memory");
        }
        __syncthreads();

        // ---- compute: 8 K-steps of 4; per step 8 frag loads -> 16 WMMAs
#pragma unroll
        for (int kk = 0; kk < 32; kk += 4) {
            v2f a[4], bf[4];
#pragma unroll
            for (int i = 0; i < 4; ++i) {
                a[i].x = As[buf][wm * 64 + i * 16 + l16][kk + 2 * half];
                a[i].y = As[buf][wm * 64 + i * 16 + l16][kk + 2 * half + 1];
            }
#pragma unroll
            for (int j = 0; j < 4; ++j) {
                bf[j].x = Bs[buf][wn * 64 + j * 16 + l16][kk + 2 * half];
                bf[j].y = Bs[buf][wn * 64 + j * 16 + l16][kk + 2 * half + 1];
            }
#pragma unroll
            for (int i = 0; i < 4; ++i)
#pragma unroll
                for (int j = 0; j < 4; ++j)
                    c[i][j] = __builtin_amdgcn_wmma_f32_16x16x4_f32(
                        false, a[i], false, bf[j], (short)0, c[i][j], false, false);
        }
        __syncthreads();                 // buffer free for tile k0+64
    }

    // ---- scatter C: frag VGPR v holds row (v + 8*half), col l16
    int mb = wm * 64 + 8 * half;
#pragma unroll
    for (int i = 0; i < 4; ++i) {
#pragma unroll
        for (int j = 0; j < 4; ++j) {
            int n = n0 + wn * 64 + j * 16 + l16;
#pragma unroll
            for (int v = 0; v < 8; ++v) {
                int s = rowTok[mb + i * 16 + v];
                out[((size_t)b * S_ + s) * D_ + n] = c[i][j][v];
            }
        }
    }
}

// ---------------------------------------------------------------------------
extern "C" void kernel_launch(void* const* d_in, const int* in_sizes, int n_in,
                              void* d_out, int out_size, void* d_ws, size_t ws_size,
                              hipStream_t stream) {
    const float* x  = (const float*)d_in[0];   // [B,S,D] fp32
    const float* Wb = (const float*)d_in[1];   // [D,D]
    const float* Wr = (const float*)d_in[2];   // [1,D]
    // d_in[3] = top_k (known constant K_=2048)
    float* out = (float*)d_out;

    char*  ws     = (char*)d_ws;
    float* logits = (float*)ws;                               // B*S floats
    int*   idx    = (int*)(ws + B_ * S_ * sizeof(float));     // B*K ints
    unsigned char* flags = (unsigned char*)(ws + B_ * S_ * sizeof(float)
                                               + B_ * K_ * sizeof(int)); // B*S bytes

    mod_router<<<(B_ * S_ * 32) / 256, 256, 0, stream>>>(x, Wr, logits);
    mod_select<<<dim3(S_ / 256, B_), 256, 0, stream>>>(logits, idx, flags);
    mod_copy<<<B_ * S_, 64, 0, stream>>>(x, flags, out);
    mod_gemm<<<dim3((B_ * K_) / 128, D_ / 256), 256, 0, stream>>>(x, Wb, idx, out);
}